// GATConv_test_dgl_25494925869249
// MI455X (gfx1250) — compile-verified
//
#include <hip/hip_runtime.h>
#include <stdint.h>

// GATConv forward for MI455X (gfx1250, wave32).
// Pipeline: [init] -> [WMMA fp32 GEMM ft=feat@W, A staged via async-to-LDS]
//        -> [el/er dots] -> [edge atomic-max] -> [edge exp + atomic denom]
//        -> [edge aggregate]

#define IN_FEATS   256
#define NUM_HEADS  4
#define OUT_FEATS  64
#define HF         (NUM_HEADS * OUT_FEATS)   // 256
#define NEG_SLOPE  0.2f
#define LDS_STRIDE 260   // floats per LDS row; %64==4 -> bank-conflict-free A reads

typedef __attribute__((ext_vector_type(2))) float v2f;
typedef __attribute__((ext_vector_type(8))) float v8f;

// ---------------- order-preserving float<->u32 encoding for atomicMax -------
__device__ __forceinline__ unsigned enc_f32(float f) {
    unsigned u = __float_as_uint(f);
    return (u & 0x80000000u) ? ~u : (u ^ 0x80000000u);
}
__device__ __forceinline__ float dec_f32(unsigned u) {
    unsigned b = (u & 0x80000000u) ? (u ^ 0x80000000u) : ~u;
    return __uint_as_float(b);
}

// ---------------- init: zero out + denom, m = enc(-inf) ---------------------
__global__ void gat_init_kernel(float* __restrict__ out, long out_n,
                                float* __restrict__ denom,
                                unsigned* __restrict__ m, int nh) {
    long stride = (long)gridDim.x * blockDim.x;
    for (long i = (long)blockIdx.x * blockDim.x + threadIdx.x; i < out_n; i += stride)
        out[i] = 0.0f;
    for (long i = (long)blockIdx.x * blockDim.x + threadIdx.x; i < nh; i += stride) {
        denom[i] = 0.0f;
        m[i] = 0x007FFFFFu;  // enc(-inf)
    }
}

// ---------------- GEMM: ft[N,256] = feat[N,256] @ W[256,256] via fp32 WMMA --
// Block = 256 threads (8 waves). Block b owns rows [16b, 16b+16).
// The 16x256 A tile is staged into LDS once per block with
// global_load_async_to_lds_b128 (ASYNCcnt), then each wave computes a 16x32
// output strip (2 col tiles, one shared A fetch per K step, 2 WMMAs).
// Requires N % 16 == 0 (50000 = 3125*16).
__global__ __launch_bounds__(256) void gat_gemm_wmma_kernel(
        const float* __restrict__ feat, const float* __restrict__ W,
        float* __restrict__ ft) {
    __shared__ float ldsA[16][LDS_STRIDE];

    const int row0 = blockIdx.x * 16;
    const int tid  = threadIdx.x;
    const int wave = tid >> 5;
    const int lane = tid & 31;
    const int l = lane & 15;   // M index for A / N index for B,D
    const int h = lane >> 4;   // lane-half: selects K pair (A,B) / +8 rows (D)

    // ---- async stage of A tile: thread tid copies 64B of row (tid/16) -----
    {
        const int r = tid >> 4;                   // 0..15 tile row
        const int c = tid & 15;                   // 0..15 64-byte chunk
        unsigned lds_off = (unsigned)(uintptr_t)&ldsA[r][c * 16];
        unsigned long long ga =
            (unsigned long long)(uintptr_t)(feat + (size_t)(row0 + r) * IN_FEATS + c * 16);
        // INST_OFFSET applies to both LDS and memory addresses (GV mode).
        asm volatile(
            "global_load_async_to_lds_b128 %0, %1, off\n\t"
            "global_load_async_to_lds_b128 %0, %1, off offset:16\n\t"
            "global_load_async_to_lds_b128 %0, %1, off offset:32\n\t"
            "global_load_async_to_lds_b128 %0, %1, off offset:48"
            :: "v"(lds_off), "v"(ga) : "memory");
        asm volatile("s_wait_asynccnt 0x0" ::: "memory");
    }
    __syncthreads();

    // ---- compute: wave w owns cols [32w, 32w+32) ---------------------------
    const int col0 = wave * 32;
    v8f acc0 = {}, acc1 = {};
    const float* arow = &ldsA[l][2 * h];                       // LDS, 8B aligned
    const float* bcol = W + (size_t)(2 * h) * HF + col0 + l;   // global

    #pragma unroll 4
    for (int k0 = 0; k0 < IN_FEATS; k0 += 4) {
        // A 16x4 fp32: vgpr j holds K = k0 + 2h + j, row = l  (ds_load_b64)
        v2f a = *(const v2f*)(arow + k0);
        // B 4x16 fp32 (two adjacent col tiles): vgpr j holds K = k0+2h+j, col = l
        v2f b0, b1;
        b0.x = bcol[(size_t)k0 * HF];
        b0.y = bcol[(size_t)(k0 + 1) * HF];
        b1.x = bcol[(size_t)k0 * HF + 16];
        b1.y = bcol[(size_t)(k0 + 1) * HF + 16];
        acc0 = __builtin_amdgcn_wmma_f32_16x16x4_f32(
            false, a, false, b0, (short)0, acc0, false, false);
        acc1 = __builtin_amdgcn_wmma_f32_16x16x4_f32(
            false, a, false, b1, (short)0, acc1, false, false);
    }

    // D 16x16 fp32: vgpr r -> row (r + 8h), col l
    #pragma unroll
    for (int r = 0; r < 8; ++r) {
        ft[(size_t)(row0 + r + 8 * h) * HF + col0 + l]      = acc0[r];
        ft[(size_t)(row0 + r + 8 * h) * HF + col0 + 16 + l] = acc1[r];
    }
}

// ---------------- el/er: per (node, head) 64-length dot ---------------------
__global__ void gat_attn_logits_kernel(const float* __restrict__ ft,
                                       const float* __restrict__ attn_l,
                                       const float* __restrict__ attn_r,
                                       float* __restrict__ el,
                                       float* __restrict__ er, int N) {
    int i = blockIdx.x * blockDim.x + threadIdx.x;
    if (i >= N * NUM_HEADS) return;
    const int n = i >> 2, hh = i & 3;
    const float* fp = ft + (size_t)n * HF + hh * OUT_FEATS;
    const float* al = attn_l + hh * OUT_FEATS;
    const float* ar = attn_r + hh * OUT_FEATS;
    float sl = 0.f, sr = 0.f;
    #pragma unroll 8
    for (int f = 0; f < OUT_FEATS; ++f) {
        float v = fp[f];
        sl += v * al[f];
        sr += v * ar[f];
    }
    el[i] = sl;
    er[i] = sr;
}

// ---------------- segment max over dst (encoded u32 atomicMax) --------------
__global__ void gat_edge_max_kernel(const int* __restrict__ src,
                                    const int* __restrict__ dst,
                                    const float* __restrict__ el,
                                    const float* __restrict__ er,
                                    unsigned* __restrict__ m, int E) {
    int i = blockIdx.x * blockDim.x + threadIdx.x;
    if (i >= E * NUM_HEADS) return;
    const int e = i >> 2, hh = i & 3;
    const int s = src[e], d = dst[e];
    float v = el[s * NUM_HEADS + hh] + er[d * NUM_HEADS + hh];
    v = (v > 0.f) ? v : NEG_SLOPE * v;
    atomicMax(&m[d * NUM_HEADS + hh], enc_f32(v));
}

// ---------------- denom: segment sum of exp(e - m[dst]) ---------------------
__global__ void gat_edge_denom_kernel(const int* __restrict__ src,
                                      const int* __restrict__ dst,
                                      const float* __restrict__ el,
                                      const float* __restrict__ er,
                                      const unsigned* __restrict__ m,
                                      float* __restrict__ denom, int E) {
    int i = blockIdx.x * blockDim.x + threadIdx.x;
    if (i >= E * NUM_HEADS) return;
    const int e = i >> 2, hh = i & 3;
    const int s = src[e], d = dst[e];
    const int dh = d * NUM_HEADS + hh;
    float v = el[s * NUM_HEADS + hh] + er[dh];
    v = (v > 0.f) ? v : NEG_SLOPE * v;
    atomicAdd(&denom[dh], __expf(v - dec_f32(m[dh])));
}

// ---------------- aggregate: out[dst] += ft[src] * a ------------------------
// One 256-thread block per edge. thread t -> head t/64, feature t%64.
// A wave (32 lanes) spans one head, so lane 0 computes the softmax weight
// once and broadcasts it (wave32 shuffle).
__global__ __launch_bounds__(256) void gat_edge_aggregate_kernel(
        const int* __restrict__ src, const int* __restrict__ dst,
        const float* __restrict__ el, const float* __restrict__ er,
        const unsigned* __restrict__ m, const float* __restrict__ denom,
        const float* __restrict__ ft, float* __restrict__ out, int E) {
    const int e = blockIdx.x;
    if (e >= E) return;
    const int t = threadIdx.x;          // 0..255
    const int hh = t >> 6;              // head
    const int s = src[e], d = dst[e];
    float a = 0.f;
    if ((t & 31) == 0) {
        const int dh = d * NUM_HEADS + hh;
        float v = el[s * NUM_HEADS + hh] + er[dh];
        v = (v > 0.f) ? v : NEG_SLOPE * v;
        a = __expf(v - dec_f32(m[dh])) / denom[dh];
    }
    a = __shfl(a, 0);                   // wave32 broadcast
    atomicAdd(&out[(size_t)d * HF + t], ft[(size_t)s * HF + t] * a);
}

// ---------------------------------------------------------------------------
extern "C" void kernel_launch(void* const* d_in, const int* in_sizes, int n_in,
                              void* d_out, int out_size, void* d_ws, size_t ws_size,
                              hipStream_t stream) {
    const float* feat   = (const float*)d_in[0];
    const int*   src    = (const int*)  d_in[1];
    const int*   dst    = (const int*)  d_in[2];
    const float* W      = (const float*)d_in[3];
    const float* attn_l = (const float*)d_in[4];
    const float* attn_r = (const float*)d_in[5];
    float* out = (float*)d_out;

    const int N = in_sizes[0] / IN_FEATS;   // 50000
    const int E = in_sizes[1];              // 800000
    const int NH = N * NUM_HEADS;

    // Workspace layout (~54.4 MB): ft | el | er | m(u32) | denom
    float*    ft    = (float*)d_ws;
    float*    el    = ft + (size_t)N * HF;
    float*    er    = el + NH;
    unsigned* m     = (unsigned*)(er + NH);
    float*    denom = (float*)(m + NH);

    // 1) init outputs/accumulators
    {
        long out_n = (long)N * HF;
        gat_init_kernel<<<2048, 256, 0, stream>>>(out, out_n, denom, m, NH);
    }
    // 2) ft = feat @ W via fp32 WMMA (N is a multiple of 16)
    gat_gemm_wmma_kernel<<<N / 16, 256, 0, stream>>>(feat, W, ft);
    // 3) el/er
    gat_attn_logits_kernel<<<(NH + 255) / 256, 256, 0, stream>>>(ft, attn_l, attn_r, el, er, N);
    // 4) segment max
    gat_edge_max_kernel<<<(E * NUM_HEADS + 255) / 256, 256, 0, stream>>>(src, dst, el, er, m, E);
    // 5) denom
    gat_edge_denom_kernel<<<(E * NUM_HEADS + 255) / 256, 256, 0, stream>>>(src, dst, el, er, m, denom, E);
    // 6) aggregate
    gat_edge_aggregate_kernel<<<E, 256, 0, stream>>>(src, dst, el, er, m, denom, ft, out, E);
}